// QuantizedLinear_57672820850769
// MI455X (gfx1250) — compile-verified
//
#include <hip/hip_runtime.h>
#include <stdint.h>

// ---------------------------------------------------------------------------
// QuantizedLinear for MI455X (gfx1250, wave32)
//   out[M,N] = x[M,K] @ dequant(W[N,K])^T + bias[N]
//   M=8192, K=4096, N=16384, fp32 end-to-end (exact vs reference; HBM-bound:
//   ~1.15 GB mandatory traffic / 23.3 TB/s ~= 49us >> f32-WMMA compute time).
// Pass 1: abs-max reduction over W -> scale in workspace.
// Pass 2: tiled GEMM with V_WMMA_F32_16X16X4_F32.
//   - x tile (pure copy): GLOBAL_LOAD_ASYNC_TO_LDS_B128 double-buffered
//     (ASYNCcnt path), falling back to direct global->LDS if the builtin is
//     unavailable on this toolchain.
//   - W tile: register-staged with fused quant->dequant (needs VALU anyway).
// ---------------------------------------------------------------------------

#define AS1 __attribute__((address_space(1)))
#define AS3 __attribute__((address_space(3)))

#if defined(__has_builtin)
#if __has_builtin(__builtin_amdgcn_global_load_async_to_lds_b128) && \
    __has_builtin(__builtin_amdgcn_s_wait_asynccnt)
#define HAVE_ASYNC_LDS 1
#endif
#endif
#ifndef HAVE_ASYNC_LDS
#define HAVE_ASYNC_LDS 0
#endif

typedef float v2f __attribute__((ext_vector_type(2)));
typedef float v8f __attribute__((ext_vector_type(8)));
typedef int   v4i __attribute__((vector_size(16)));   // builtin's pointee type

static constexpr int M_TOT = 8192;    // 4 * 2048
static constexpr int N_TOT = 16384;
static constexpr int K_TOT = 4096;

static constexpr int BM = 128;
static constexpr int BN = 128;
static constexpr int BK = 32;
static constexpr int LDA = BK + 4;    // padded LDS row stride (floats) for A
static constexpr int LDB = BN + 4;    // padded LDS row stride (floats) for B

// ---------------------------------------------------------------------------
// Pass 1: abs-max over the weight tensor (67.1M floats), uint-bit atomicMax.
// ---------------------------------------------------------------------------
__global__ __launch_bounds__(256) void absmax_kernel(
    const float* __restrict__ w, unsigned int* __restrict__ amax_bits, int n4) {
  __shared__ float red[256];
  const float4* w4 = (const float4*)w;
  float m = 0.0f;
  for (int i = blockIdx.x * blockDim.x + threadIdx.x; i < n4;
       i += gridDim.x * blockDim.x) {
    float4 v = w4[i];
    m = fmaxf(m, fmaxf(fmaxf(fabsf(v.x), fabsf(v.y)),
                       fmaxf(fabsf(v.z), fabsf(v.w))));
  }
  red[threadIdx.x] = m;
  __syncthreads();
#pragma unroll
  for (int s = 128; s > 0; s >>= 1) {
    if ((int)threadIdx.x < s)
      red[threadIdx.x] = fmaxf(red[threadIdx.x], red[threadIdx.x + s]);
    __syncthreads();
  }
  if (threadIdx.x == 0) {
    // |w| >= 0 so float ordering == uint ordering on the bit patterns.
    atomicMax(amax_bits, __float_as_uint(red[0]));
  }
}

// ---------------------------------------------------------------------------
// Pass 2: WMMA f32 GEMM. Block = 128x128, 8 waves; wave = 64x32 (4x2 tiles of
// 16x16). A tile async-copied to double-buffered LDS; B tile reg-prefetched
// with fused dequant.
// ---------------------------------------------------------------------------
__global__ __launch_bounds__(256, 1) void qlinear_gemm_kernel(
    const float* __restrict__ x, const float* __restrict__ w,
    const float* __restrict__ bias, float* __restrict__ out,
    const unsigned int* __restrict__ amax_bits) {
  __shared__ float As[2][BM][LDA];  // 2 * 128*36*4 = 36.9 KB, x tile [m][k]
  __shared__ float Bs[BK][LDB];     //     32*132*4 = 16.9 KB, W tile [k][n]

  const float amax = __uint_as_float(*amax_bits);
  const float scale = amax * (1.0f / 127.0f);
  const float inv_scale = (amax > 0.0f) ? (127.0f / amax) : 0.0f;

  const int tid  = threadIdx.x;
  const int lane = tid & 31;
  const int wave = tid >> 5;
  const int lm   = lane & 15;   // M/N index within a 16-wide fragment
  const int hi   = lane >> 4;   // half-wave (selects K pair / row+8)

  const int bm = blockIdx.x * BM;   // M fastest -> x strip stays L2 resident
  const int bn = blockIdx.y * BN;
  const int wm = (wave >> 2) * 64;  // 2 waves along M
  const int wn = (wave & 3) * 32;   // 4 waves along N

  // Per-thread tile mapping: 1024 float4 per 128x32 tile / 256 thr = 4 each.
  int arow[4], acol[4];
#pragma unroll
  for (int i = 0; i < 4; ++i) {
    int idx = tid + i * 256;
    arow[i] = idx >> 3;             // 0..127
    acol[i] = idx & 7;              // 0..7 (float4 index along K)
  }

  // ---- A tile copy: async global->LDS when available --------------------
  auto copy_a = [&](int buf, int k0) {
#if HAVE_ASYNC_LDS
#pragma unroll
    for (int i = 0; i < 4; ++i) {
      // Two-step cast: reinterpret in generic AS, then pure addrspacecast.
      v4i* gp = reinterpret_cast<v4i*>(const_cast<float*>(
          &x[(size_t)(bm + arow[i]) * K_TOT + k0 + acol[i] * 4]));
      v4i* lp = reinterpret_cast<v4i*>(&As[buf][arow[i]][acol[i] * 4]);
      __builtin_amdgcn_global_load_async_to_lds_b128(
          (AS1 v4i*)gp, (AS3 v4i*)lp, /*imm_offset=*/0, /*cpol=*/0);
    }
#else
#pragma unroll
    for (int i = 0; i < 4; ++i) {
      float4 t =
          *(const float4*)&x[(size_t)(bm + arow[i]) * K_TOT + k0 + acol[i] * 4];
      *(float4*)&As[buf][arow[i]][acol[i] * 4] = t;
    }
#endif
  };
  auto wait_a = [&]() {
#if HAVE_ASYNC_LDS
    __builtin_amdgcn_s_wait_asynccnt(0);
#endif
  };

  // ---- B tile: global->regs with fused quant->dequant, then LDS ---------
  float4 bReg[4];
  auto load_b = [&](int k0) {
#pragma unroll
    for (int i = 0; i < 4; ++i)
      bReg[i] =
          *(const float4*)&w[(size_t)(bn + arow[i]) * K_TOT + k0 + acol[i] * 4];
  };
  auto commit_b = [&]() {
#pragma unroll
    for (int i = 0; i < 4; ++i) {
      float q0 = fminf(fmaxf(rintf(bReg[i].x * inv_scale), -128.0f), 127.0f);
      float q1 = fminf(fmaxf(rintf(bReg[i].y * inv_scale), -128.0f), 127.0f);
      float q2 = fminf(fmaxf(rintf(bReg[i].z * inv_scale), -128.0f), 127.0f);
      float q3 = fminf(fmaxf(rintf(bReg[i].w * inv_scale), -128.0f), 127.0f);
      Bs[acol[i] * 4 + 0][arow[i]] = q0 * scale;   // stored transposed [k][n]
      Bs[acol[i] * 4 + 1][arow[i]] = q1 * scale;
      Bs[acol[i] * 4 + 2][arow[i]] = q2 * scale;
      Bs[acol[i] * 4 + 3][arow[i]] = q3 * scale;
    }
  };

  const int ntiles = K_TOT / BK;    // 128

  // --- prologue: tile 0 ---
  copy_a(0, 0);
  load_b(0);
  commit_b();
  wait_a();
  __syncthreads();

  v8f acc[4][2] = {};

  for (int kt = 0; kt < ntiles; ++kt) {
    const int cur = kt & 1;

    // Issue next A tile into the other buffer + prefetch next B into regs.
    if (kt + 1 < ntiles) {
      copy_a(1 - cur, (kt + 1) * BK);
      load_b((kt + 1) * BK);
    }

    // --- compute on current LDS tiles ---
#pragma unroll
    for (int kk = 0; kk < BK; kk += 4) {
      v2f a[4];
      v2f b[2];
#pragma unroll
      for (int mi = 0; mi < 4; ++mi) {
        // A 16x4 f32: lane holds row m=lane%16; VGPR0/1 = K = 2*hi + {0,1}.
        a[mi] = *(const v2f*)&As[cur][wm + mi * 16 + lm][kk + 2 * hi];
      }
#pragma unroll
      for (int ni = 0; ni < 2; ++ni) {
        // B 4x16 f32: N striped across lanes; VGPR v + half hi -> K = v+2*hi.
        const int n = wn + ni * 16 + lm;
        b[ni].x = Bs[kk + 2 * hi + 0][n];
        b[ni].y = Bs[kk + 2 * hi + 1][n];
      }
#pragma unroll
      for (int mi = 0; mi < 4; ++mi) {
#pragma unroll
        for (int ni = 0; ni < 2; ++ni) {
          acc[mi][ni] = __builtin_amdgcn_wmma_f32_16x16x4_f32(
              /*neg_a=*/false, a[mi], /*neg_b=*/false, b[ni],
              /*c_mod=*/(short)0, acc[mi][ni],
              /*reuse_a=*/false, /*reuse_b=*/false);
        }
      }
    }

    // All my async A writes done, then everyone past their reads of Bs/As.
    wait_a();
    __syncthreads();

    if (kt + 1 < ntiles) {
      commit_b();
      __syncthreads();
    }
  }

  // --- epilogue: C/D 16x16 f32 layout: VGPR r -> row r (lanes 0-15) or
  // r+8 (lanes 16-31), N = lane%16. NT stores keep the 512MB output stream
  // from evicting the L2-resident x. ---
#pragma unroll
  for (int ni = 0; ni < 2; ++ni) {
    const int n = bn + wn + ni * 16 + lm;
    const float bv = bias[n];
#pragma unroll
    for (int mi = 0; mi < 4; ++mi) {
#pragma unroll
      for (int r = 0; r < 8; ++r) {
        const int m = bm + wm + mi * 16 + r + 8 * hi;
        __builtin_nontemporal_store(acc[mi][ni][r] + bv,
                                    &out[(size_t)m * N_TOT + n]);
      }
    }
  }
}

// ---------------------------------------------------------------------------
extern "C" void kernel_launch(void* const* d_in, const int* in_sizes, int n_in,
                              void* d_out, int out_size, void* d_ws,
                              size_t ws_size, hipStream_t stream) {
  const float* x    = (const float*)d_in[0];   // [4,2048,4096]
  const float* w    = (const float*)d_in[1];   // [16384,4096]
  const float* bias = (const float*)d_in[2];   // [16384]
  float* out = (float*)d_out;                  // [4,2048,16384]

  unsigned int* amax_bits = (unsigned int*)d_ws;
  (void)hipMemsetAsync(amax_bits, 0, sizeof(unsigned int), stream);

  absmax_kernel<<<1024, 256, 0, stream>>>(w, amax_bits, (N_TOT * K_TOT) / 4);

  dim3 grid(M_TOT / BM, N_TOT / BN);   // M fastest -> x stays in 192MB L2
  qlinear_gemm_kernel<<<grid, 256, 0, stream>>>(x, w, bias, out, amax_bits);
}